// Depthwise_80719615361306
// MI455X (gfx1250) — compile-verified
//
#include <hip/hip_runtime.h>
#include <cstdint>

// Problem geometry (fixed by the reference).
constexpr int kN      = 32;
constexpr int kC      = 128;
constexpr int kH      = 112;
constexpr int kW      = 112;
constexpr int kImgs   = kN * kC;      // 4096 channel-images
constexpr int kTileH  = 16;           // output rows per block
constexpr int kTiles  = kH / kTileH;  // 7
constexpr int kRows   = kTileH + 2;   // 18 LDS rows (with halo)
constexpr int kPitch  = 120;          // floats; data at [4..115+4], halos at 3 and 116
constexpr int kDataOfs= 4;            // 16B-aligned start of row data in LDS
constexpr int kChunks = kW / 4;       // 28 x b128 chunks per row
constexpr int kBlock  = 224;          // 7 wave32s; tid%112 -> column

#if defined(__gfx1250__) && __has_builtin(__builtin_amdgcn_global_load_async_to_lds_b128)
#define ASYNC_MODE 2
#elif defined(__gfx1250__) && __has_builtin(__builtin_amdgcn_global_load_async_to_lds_b32)
#define ASYNC_MODE 1
#else
#define ASYNC_MODE 0
#endif

__device__ __forceinline__ float fsign(float v) {
    // jnp.sign semantics for finite inputs: (-1, 0, +1)
    return (float)((v > 0.0f) - (v < 0.0f));
}

#if ASYNC_MODE
// b32 builtin takes addrspace-qualified scalar int pointers; b128 takes
// 4-wide int vector pointers (per hipcc diagnostics).
typedef int v4i __attribute__((ext_vector_type(4)));
typedef __attribute__((address_space(1))) int*  g32_ptr;
typedef __attribute__((address_space(3))) int*  l32_ptr;
typedef __attribute__((address_space(1))) v4i* g128_ptr;
typedef __attribute__((address_space(3))) v4i* l128_ptr;

// Flat AMDGPU global addresses are identity-mapped -> reinterpret via integer.
__device__ __forceinline__ g32_ptr  to_global32(const void* p) {
    return (g32_ptr)(unsigned long long)(uintptr_t)p;
}
__device__ __forceinline__ g128_ptr to_global128(const void* p) {
    return (g128_ptr)(unsigned long long)(uintptr_t)p;
}
// Flat LDS address keeps the LDS offset in the low 32 bits (ISA 10.2) -> truncate.
__device__ __forceinline__ l32_ptr  to_lds32(void* p) {
    return (l32_ptr)(uint32_t)(uintptr_t)p;
}
__device__ __forceinline__ l128_ptr to_lds128(void* p) {
    return (l128_ptr)(uint32_t)(uintptr_t)p;
}
__device__ __forceinline__ void wait_async0() {
#if __has_builtin(__builtin_amdgcn_s_wait_asynccnt)
    __builtin_amdgcn_s_wait_asynccnt(0);
#else
    asm volatile("s_wait_asynccnt 0" ::: "memory");
#endif
}
#endif

__global__ __launch_bounds__(kBlock)
void bnn_dwconv3x3_fused(const float* __restrict__ x,
                         const float* __restrict__ wgt,
                         float* __restrict__ out) {
    const int bid  = blockIdx.x;
    const int img  = bid / kTiles;          // n*C + c
    const int tile = bid - img * kTiles;
    const int c    = img & (kC - 1);
    const int h0   = tile * kTileH;
    const int tid  = threadIdx.x;

    __shared__ float smem[kRows * kPitch];  // 8640 B

    // Per-channel 3x3 weights + their signs (uniform across the block -> SGPRs).
    float wt[9], swt[9];
    const float* wp = wgt + c * 9;
#pragma unroll
    for (int t = 0; t < 9; ++t) { wt[t] = wp[t]; swt[t] = fsign(wt[t]); }

    // Zero only the halo cells that are read but never loaded:
    //  - column indices 3 (x[-1]) and 116 (x[112]) of every tile row
    //  - the top halo row when h0 == 0, the bottom halo row for the last tile
    for (int r = tid; r < kRows; r += kBlock) {
        smem[r * kPitch + 3]   = 0.0f;
        smem[r * kPitch + 116] = 0.0f;
    }
    if (h0 == 0) {
        for (int i = tid; i < kW; i += kBlock) smem[0 * kPitch + kDataOfs + i] = 0.0f;
    }
    if (h0 + kTileH >= kH) {
        for (int i = tid; i < kW; i += kBlock) smem[(kRows - 1) * kPitch + kDataOfs + i] = 0.0f;
    }
    __syncthreads();

    // Stage the in-bounds rows (+halo rows) of x into LDS.
    const float* src    = x + (size_t)img * (kH * kW);
    const int    row_lo = (h0 == 0) ? 0 : (h0 - 1);
    const int    row_hi = (h0 + kTileH >= kH) ? (kH - 1) : (h0 + kTileH);
    const int    nrows  = row_hi - row_lo + 1;

#if ASYNC_MODE == 2
    // 16B chunks: global row base and LDS data offset are both 16B aligned.
    for (int i = tid; i < nrows * kChunks; i += kBlock) {
        const int rr = i / kChunks;
        const int ch = i - rr * kChunks;
        const int gh = row_lo + rr;
        const int lr = gh - (h0 - 1);               // tile row in [0, 17]
        __builtin_amdgcn_global_load_async_to_lds_b128(
            to_global128(src + gh * kW + ch * 4),
            to_lds128(&smem[lr * kPitch + kDataOfs + ch * 4]),
            /*offset=*/0, /*cpol=*/0);
    }
    wait_async0();
#elif ASYNC_MODE == 1
    for (int i = tid; i < nrows * kW; i += kBlock) {
        const int rr = i / kW;
        const int cc = i - rr * kW;
        const int gh = row_lo + rr;
        const int lr = gh - (h0 - 1);
        __builtin_amdgcn_global_load_async_to_lds_b32(
            to_global32(src + gh * kW + cc),
            to_lds32(&smem[lr * kPitch + kDataOfs + cc]),
            /*offset=*/0, /*cpol=*/0);
    }
    wait_async0();
#else
    for (int i = tid; i < nrows * kW; i += kBlock) {
        const int rr = i / kW;
        const int cc = i - rr * kW;
        const int gh = row_lo + rr;
        const int lr = gh - (h0 - 1);
        smem[lr * kPitch + kDataOfs + cc] = src[gh * kW + cc];
    }
#endif
    __syncthreads();

    // Compute: each thread owns one column and an 8-row vertical run,
    // sliding a 3x3 register window (3 new LDS reads per output row).
    const int col = tid % kW;        // 0..111
    const int rg  = tid / kW;        // 0..1
    float* o = out + (size_t)img * (kH * kW);

    // Taps for output col are smem columns col+3, col+4, col+5.
    const float* base = &smem[col + 3];
    const int lh = rg * 8 + 1;       // tile row of first output row

    float m0 = base[(lh - 1) * kPitch + 0];
    float m1 = base[(lh - 1) * kPitch + 1];
    float m2 = base[(lh - 1) * kPitch + 2];
    float z0 = base[(lh    ) * kPitch + 0];
    float z1 = base[(lh    ) * kPitch + 1];
    float z2 = base[(lh    ) * kPitch + 2];

#pragma unroll
    for (int i = 0; i < 8; ++i) {
        const float* pr = base + (lh + 1 + i) * kPitch;
        const float p0 = pr[0];
        const float p1 = pr[1];
        const float p2 = pr[2];

        // out = sum_t sign(x_t)*w_t + x_t*sign(w_t)
        float acc = 0.0f;
        acc = fmaf(fsign(m0), wt[0], fmaf(m0, swt[0], acc));
        acc = fmaf(fsign(m1), wt[1], fmaf(m1, swt[1], acc));
        acc = fmaf(fsign(m2), wt[2], fmaf(m2, swt[2], acc));
        acc = fmaf(fsign(z0), wt[3], fmaf(z0, swt[3], acc));
        acc = fmaf(fsign(z1), wt[4], fmaf(z1, swt[4], acc));
        acc = fmaf(fsign(z2), wt[5], fmaf(z2, swt[5], acc));
        acc = fmaf(fsign(p0), wt[6], fmaf(p0, swt[6], acc));
        acc = fmaf(fsign(p1), wt[7], fmaf(p1, swt[7], acc));
        acc = fmaf(fsign(p2), wt[8], fmaf(p2, swt[8], acc));

        const int h = h0 + rg * 8 + i;
        __builtin_nontemporal_store(acc, &o[h * kW + col]);

        m0 = z0; m1 = z1; m2 = z2;
        z0 = p0; z1 = p1; z2 = p2;
    }
}

extern "C" void kernel_launch(void* const* d_in, const int* in_sizes, int n_in,
                              void* d_out, int out_size, void* d_ws, size_t ws_size,
                              hipStream_t stream) {
    const float* x = (const float*)d_in[0];   // [32,128,112,112] fp32
    const float* w = (const float*)d_in[1];   // [128,1,3,3] fp32
    // d_in[2], d_in[3] = alpha_x, alpha_w: only affect STE backward, unused in fwd.
    float* out = (float*)d_out;               // [32,128,112,112] fp32

    dim3 grid(kImgs * kTiles);                // 28672 blocks
    dim3 block(kBlock);                       // 7 wave32s
    bnn_dwconv3x3_fused<<<grid, block, 0, stream>>>(x, w, out);
}